// ROIRotate_56959856279976
// MI455X (gfx1250) — compile-verified
//
#include <hip/hip_runtime.h>
#include <hip/hip_bf16.h>
#include <stdint.h>

#define NB     1024   // N*K boxes
#define CCH    32     // channels
#define FH     128
#define FW     128
#define OH     8      // HEIGHT
#define OW     64     // MAX_W
#define TILE_R 18     // max source rows needed per box (bh<=15 + jitter + tap)
#define TILE_C 56     // max source cols needed per box (bw<=50 + jitter + tap)
#define CPASS  16     // channels per TDM pass (16*18*56*4B = 63KB LDS)

typedef __attribute__((ext_vector_type(4))) unsigned int u32x4;
typedef __attribute__((ext_vector_type(8))) int i32x8;
typedef __attribute__((ext_vector_type(4))) int i32x4;

__device__ __forceinline__ int imin(int a, int b) { return a < b ? a : b; }
__device__ __forceinline__ int imax(int a, int b) { return a > b ? a : b; }

// ---------------- kernel 1: per-box affine params + width ----------------
__global__ void roi_params_kernel(const float* __restrict__ boxes,
                                  float* __restrict__ ws) {
  int b = blockIdx.x * blockDim.x + threadIdx.x;
  if (b >= NB) return;
  const float* bx = boxes + b * 8;
  float x1 = bx[0] * 0.25f, y1 = bx[1] * 0.25f;
  float x2 = bx[2] * 0.25f, y2 = bx[3] * 0.25f;
  float x4 = bx[6] * 0.25f, y4 = bx[7] * 0.25f;
  float bw = x2 - x1, bh = y4 - y1;
  int   w  = (int)(8.0f * bw / bh);   // trunc (positive)
  float wf = (float)w;
  ws[0 * NB + b] = bw / wf;              // A00
  ws[1 * NB + b] = (y2 - y1) / wf;       // A10
  ws[2 * NB + b] = (x4 - x1) * 0.125f;   // A01
  ws[3 * NB + b] = bh * 0.125f;          // A11
  ws[4 * NB + b] = x1;                   // tx
  ws[5 * NB + b] = y1;                   // ty
  ((int*)ws)[6 * NB + b] = w;            // width
}

// ---------------- kernel 2: stable descending sort by width ----------------
// key = (width<<10)|idx ascending == stable ascending argsort; then reverse.
__global__ void roi_sort_kernel(float* __restrict__ ws,
                                float* __restrict__ out_tail) {
  __shared__ int keys[NB];
  const int tid = threadIdx.x;
  int w = ((const int*)ws)[6 * NB + tid];
  keys[tid] = (w << 10) | tid;
  __syncthreads();
  for (int k = 2; k <= NB; k <<= 1) {
    for (int j = k >> 1; j > 0; j >>= 1) {
      int ixj = tid ^ j;
      if (ixj > tid) {
        int a = keys[tid], b = keys[ixj];
        bool up = ((tid & k) == 0);
        if ((a > b) == up) { keys[tid] = b; keys[ixj] = a; }
      }
      __syncthreads();
    }
  }
  int rev = keys[NB - 1 - tid];               // descending view
  ((int*)ws)[7 * NB + tid] = rev & 1023;      // order for kernel 3
  out_tail[tid]      = (float)(rev >> 10);    // widths[order]
  out_tail[NB + tid] = (float)(rev & 1023);   // order
}

// ---------------- kernel 3: TDM tile -> LDS, bilinear warp ----------------
__global__ __launch_bounds__(512)
void roi_warp_kernel(const float* __restrict__ fm, const float* __restrict__ ws,
                     float* __restrict__ out) {
  __shared__ float tile[CPASS * TILE_R * TILE_C];
  const int j = blockIdx.x;                       // output slot
  const int s = ((const int*)ws)[7 * NB + j];     // source box
  const float a00 = ws[0 * NB + s], a10 = ws[1 * NB + s];
  const float a01 = ws[2 * NB + s], a11 = ws[3 * NB + s];
  const float tx  = ws[4 * NB + s], ty  = ws[5 * NB + s];
  const int   w   = ((const int*)ws)[6 * NB + s];
  const int   img = s >> 5;                       // s / K, K=32

  // Bounding box of all valid sample points (+1 for the far tap).
  const float wm1   = (float)(w - 1);
  const float a017  = a01 * 7.0f;
  const float a10w  = a10 * wm1;
  const float minsx = tx + fminf(0.f, a017);
  const float maxsx = tx + a00 * wm1 + fmaxf(0.f, a017);
  const float minsy = ty + fminf(0.f, a10w);
  const float maxsy = ty + fmaxf(0.f, a10w) + a11 * 7.0f;
  const int ix0  = imax(0, (int)floorf(minsx));
  const int iy0  = imax(0, (int)floorf(minsy));
  const int cols = imin(TILE_C, imin(FW - ix0, (int)floorf(maxsx) + 2 - ix0));
  const int rows = imin(TILE_R, imin(FH - iy0, (int)floorf(maxsy) + 2 - iy0));
  const int plane = rows * cols;

  const int lane = threadIdx.x & 31;
  const int wv   = threadIdx.x >> 5;          // 0..15
  const int y    = wv >> 1;                   // output row 0..7
  const int x    = lane + ((wv & 1) << 5);    // output col 0..63

  float acc[CCH];
  #pragma unroll
  for (int c = 0; c < CCH; ++c) acc[c] = 0.f;

  const bool active = (x < w);
  float w00 = 0.f, w01 = 0.f, w10 = 0.f, w11 = 0.f;
  int base = 0;
  if (active) {
    float sx = a00 * (float)x + a01 * (float)y + tx;
    float sy = a10 * (float)x + a11 * (float)y + ty;
    int x0 = (int)floorf(sx), y0 = (int)floorf(sy);
    float fx = sx - (float)x0, fy = sy - (float)y0;
    int rx = imin(imax(x0 - ix0, 0), cols - 2);
    int ry = imin(imax(y0 - iy0, 0), rows - 2);
    w00 = (1.f - fy) * (1.f - fx);
    w01 = (1.f - fy) * fx;
    w10 = fy * (1.f - fx);
    w11 = fy * fx;
    base = ry * cols + rx;
  }

  #pragma unroll
  for (int p = 0; p < 2; ++p) {
    const int c0 = p * CPASS;
    __syncthreads();   // previous pass fully consumed before tile overwrite
#if __has_builtin(__builtin_amdgcn_tensor_load_to_lds)
    if (threadIdx.x == 0) {
      // 3-D tile: dim0=cols (contig x), dim1=rows (stride FW), dim2=16ch (stride FH*FW)
      uint64_t ga = (uint64_t)(uintptr_t)fm +
          4ull * ((uint64_t)(img * CCH + c0) * (FH * FW) +
                  (uint64_t)iy0 * FW + (uint64_t)ix0);
      uint32_t lds = (uint32_t)(uintptr_t)tile;  // LDS byte offset (low 32 bits)
      u32x4 g0; i32x8 g1; i32x4 g2; i32x4 g3; i32x8 g4;
      // Group0: count=1 | lds_addr | global_addr[56:0] | type=2
      g0[0] = 1u;
      g0[1] = (uint32_t)__builtin_amdgcn_readfirstlane((int)lds);
      g0[2] = (uint32_t)__builtin_amdgcn_readfirstlane((int)(uint32_t)ga);
      g0[3] = (uint32_t)__builtin_amdgcn_readfirstlane(
                  (int)(((uint32_t)(ga >> 32) & 0x1FFFFFFu) | (2u << 30)));
      // Group1: data_size=4B; tensor_dim0=128, dim1=128; tile = cols x rows x 16;
      //         dim0_stride=128, dim1_stride=16384
      g1[0] = 2 << 16;                                 // data_size=2 (4 bytes)
      g1[1] = (int)((unsigned)FW << 16);               // tensor_dim0[15:0]
      g1[2] = (int)((unsigned)FH << 16);               // dim0 hi=0 | tensor_dim1[15:0]
      g1[3] = __builtin_amdgcn_readfirstlane(cols << 16);           // dim1 hi=0 | tile_dim0
      g1[4] = __builtin_amdgcn_readfirstlane(rows | (CPASS << 16)); // tile_dim1 | tile_dim2
      g1[5] = FW;                                      // tensor_dim0_stride[31:0]
      g1[6] = (int)(((unsigned)(FH * FW) & 0xFFFFu) << 16); // d0s hi=0 | d1s[15:0]
      g1[7] = (FH * FW) >> 16;                         // tensor_dim1_stride[47:16]
      // Group2: tensor_dim2=32 | tensor_dim3=1 | tensor_dim2_stride=FH*FW | tile_dim3=0
      g2[0] = CCH; g2[1] = 1; g2[2] = FH * FW; g2[3] = 0;
      // Group3: dim3 stride / dim4 / tile_dim4 all unused
      g3[0] = 0; g3[1] = 0; g3[2] = 0; g3[3] = 0;
      // Extra group required by the 6-arg builtin on this toolchain: zero-fill.
      g4[0] = 0; g4[1] = 0; g4[2] = 0; g4[3] = 0;
      g4[4] = 0; g4[5] = 0; g4[6] = 0; g4[7] = 0;
      __builtin_amdgcn_tensor_load_to_lds(g0, g1, g2, g3, g4, 0);
      __builtin_amdgcn_s_wait_tensorcnt(0);
    }
#else
    {
      const float* gsrc = fm + (size_t)(img * CCH + c0) * (FH * FW)
                             + (size_t)iy0 * FW + (size_t)ix0;
      const int total = CPASS * plane;
      for (int e = threadIdx.x; e < total; e += 512) {
        int c  = e / plane;
        int r2 = e - c * plane;
        int rr = r2 / cols;
        int xx = r2 - rr * cols;
        tile[e] = gsrc[(size_t)c * (FH * FW) + (size_t)rr * FW + xx];
      }
    }
#endif
    __syncthreads();
    if (active) {
      #pragma unroll
      for (int c = 0; c < CPASS; ++c) {
        const float* t = tile + c * plane + base;
        acc[c0 + c] += w00 * t[0] + w01 * t[1] + w10 * t[cols] + w11 * t[cols + 1];
      }
    }
  }

  // crops[j][y][x][0..31]: contiguous 128B per lane, masked cols stay zero.
  float4* o = (float4*)(out + (((size_t)j * OH + y) * OW + x) * CCH);
  #pragma unroll
  for (int q = 0; q < 8; ++q)
    o[q] = make_float4(acc[4 * q], acc[4 * q + 1], acc[4 * q + 2], acc[4 * q + 3]);
}

// ---------------- launch ----------------
extern "C" void kernel_launch(void* const* d_in, const int* in_sizes, int n_in,
                              void* d_out, int out_size, void* d_ws, size_t ws_size,
                              hipStream_t stream) {
  const float* fm    = (const float*)d_in[0];   // (32,32,128,128) f32
  const float* boxes = (const float*)d_in[1];   // (32,32,8) f32
  float* out = (float*)d_out;
  float* ws  = (float*)d_ws;                    // 32KB used

  roi_params_kernel<<<NB / 256, 256, 0, stream>>>(boxes, ws);
  roi_sort_kernel<<<1, NB, 0, stream>>>(ws, out + (size_t)NB * OH * OW * CCH);
  roi_warp_kernel<<<NB, 512, 0, stream>>>(fm, ws, out);
}